// YOLOLoss_83021717832615
// MI455X (gfx1250) — compile-verified
//
#include <hip/hip_runtime.h>
#include <hip/hip_bf16.h>
#include <stdint.h>

// ---------------- CDNA5 async global->LDS plumbing ----------------
#define GAS __attribute__((address_space(1)))
#define LAS __attribute__((address_space(3)))

typedef int v4i __attribute__((vector_size(16)));   // b128 payload type

#ifndef __has_builtin
#define __has_builtin(x) 0
#endif

#if __has_builtin(__builtin_amdgcn_global_load_async_to_lds_b128)
// signature (from clang diagnostic): (v4i AS1*, v4i AS3*, imm offset, imm cpol)
#define ASYNC_COPY_B128(gsrc, ldst)                                          \
  __builtin_amdgcn_global_load_async_to_lds_b128(                            \
      (GAS v4i*)(uintptr_t)(gsrc),                                           \
      (LAS v4i*)(uint32_t)(uintptr_t)(ldst), 0, 0)
#else
// fallback: raw CDNA5 mnemonic (vdst = LDS byte address VGPR, vaddr = 64b addr)
#define ASYNC_COPY_B128(gsrc, ldst)                                          \
  asm volatile("global_load_async_to_lds_b128 %0, %1, off"                   \
               :: "v"((uint32_t)(uintptr_t)(ldst)),                          \
                  "v"((const void*)(gsrc)) : "memory")
#endif

#if __has_builtin(__builtin_amdgcn_s_wait_asynccnt)
#define WAIT_ASYNC(n) do { __builtin_amdgcn_s_wait_asynccnt(n);              \
                           asm volatile("" ::: "memory"); } while (0)
#else
#define WAIT_ASYNC(n) asm volatile("s_wait_asynccnt %0" :: "i"(n) : "memory")
#endif

// ---------------- problem constants ----------------
constexpr int   CELL_F     = 30;                      // floats per cell (2*5 + 20)
constexpr int   TILE_CELLS = 64;                      // cells per tile
constexpr int   TILE_F     = TILE_CELLS * CELL_F;     // 1920 floats = 7680 B
constexpr int   TILE_BYTES = TILE_F * 4;              // 7680
constexpr int   CHUNKS     = TILE_BYTES / (32 * 16);  // 15 x b128 per lane
constexpr int   WAVES      = 4;                       // waves per block (wave32)

// ---------------- per-cell YOLO loss ----------------
__device__ __forceinline__ float cell_loss(const float* P, const float* L) {
  const float s = 1.0f / 7.0f;
  const float conf = L[4];
  const bool  obj   = conf > 0.0f;
  const bool  noobj = conf == 0.0f;

  // label box 0
  const float lx = L[0], ly = L[1], lw = L[2], lh = L[3];
  const float l1x = lx * s - 0.5f * lw, l2x = lx * s + 0.5f * lw;
  const float l1y = ly * s - 0.5f * lh, l2y = ly * s + 0.5f * lh;
  const float la = lw * lh;

  float iou0, iou1;
  {
    const float px = P[0], py = P[1], pw = P[2], ph = P[3];
    const float p1x = px * s - 0.5f * pw, p2x = px * s + 0.5f * pw;
    const float p1y = py * s - 0.5f * ph, p2y = py * s + 0.5f * ph;
    float iw = fmaxf(fminf(p2x, l2x) - fmaxf(p1x, l1x), 0.0f);
    float ih = fmaxf(fminf(p2y, l2y) - fmaxf(p1y, l1y), 0.0f);
    float inter = iw * ih;
    iou0 = inter / (pw * ph + la - inter);
  }
  {
    const float px = P[5], py = P[6], pw = P[7], ph = P[8];
    const float p1x = px * s - 0.5f * pw, p2x = px * s + 0.5f * pw;
    const float p1y = py * s - 0.5f * ph, p2y = py * s + 0.5f * ph;
    float iw = fmaxf(fminf(p2x, l2x) - fmaxf(p1x, l1x), 0.0f);
    float ih = fmaxf(fminf(p2y, l2y) - fmaxf(p1y, l1y), 0.0f);
    float inter = iw * ih;
    iou1 = inter / (pw * ph + la - inter);
  }

  const int   idx     = (iou1 > iou0) ? 1 : 0;      // jnp.argmax: first max wins ties
  const float max_iou = fmaxf(iou0, iou1);

  const float pc0 = P[4], pc1 = P[9];
  const float lc0 = L[4], lc1 = L[9];
  const float noobj_l =
      noobj ? ((pc0 - lc0) * (pc0 - lc0) + (pc1 - lc1) * (pc1 - lc1)) : 0.0f;

  const float respc  = idx ? pc1 : pc0;
  float       resp_l = respc - max_iou; resp_l *= resp_l;
  const float nrespc = idx ? pc0 : pc1;
  const float nresp_l = nrespc * nrespc;

  const float* pr = P + idx * 5;
  const float* lr = L + idx * 5;
  const float dx = pr[0] - lr[0], dy = pr[1] - lr[1];
  const float xy = dx * dx + dy * dy;
  const float dw = sqrtf(pr[2]) - sqrtf(lr[2]);
  const float dh = sqrtf(pr[3]) - sqrtf(lr[3]);
  const float wh = dw * dw + dh * dh;

  float cls = 0.0f;
#pragma unroll
  for (int c = 10; c < 30; ++c) {
    const float d = P[c] - L[c];
    cls += d * d;
  }

  const float objterm = 5.0f * (xy + wh) + 2.0f * resp_l + nresp_l + cls;
  return (obj ? objterm : 0.0f) + 0.5f * noobj_l;
}

// ---------------- async tile fetch: 64 cells (7680 B) per tensor ----------------
__device__ __forceinline__ void issue_tile(const char* gp, const char* gl,
                                           uint64_t lim, float* ldsP, float* ldsL,
                                           long tile, int lane) {
  const uint64_t base = (uint64_t)tile * TILE_BYTES + (uint64_t)lane * 16;
  char* lp = (char*)ldsP + lane * 16;
  char* ll = (char*)ldsL + lane * 16;
#pragma unroll
  for (int i = 0; i < CHUNKS; ++i) {
    uint64_t off = base + (uint64_t)i * 512;   // 32 lanes * 16 B per chunk row
    if (off > lim) off = lim;                  // clamp (tail safety; full tiles in practice)
    ASYNC_COPY_B128(gp + off, lp + i * 512);
    ASYNC_COPY_B128(gl + off, ll + i * 512);
  }
}

// ---------------- main streaming kernel ----------------
__global__ void __launch_bounds__(32 * WAVES)
yolo_loss_main(const float* __restrict__ preds, const float* __restrict__ labels,
               float* __restrict__ blockPartials, long nCells, int numTiles) {
  // [wave][double-buffer][preds/labels][tile floats]
  __shared__ __align__(16) float tiles[WAVES][2][2][TILE_F];
  __shared__ float wpart[WAVES];

  const int wave = threadIdx.x >> 5;
  const int lane = threadIdx.x & 31;
  const int wgid = blockIdx.x * WAVES + wave;
  const int stride = gridDim.x * WAVES;

  const char* gp = (const char*)preds;
  const char* gl = (const char*)labels;
  const uint64_t lim = (((uint64_t)nCells * (CELL_F * 4)) - 16) & ~(uint64_t)15;

  float acc = 0.0f;
  int  buf  = 0;
  long tile = wgid;

  if (tile < (long)numTiles)
    issue_tile(gp, gl, lim, tiles[wave][0][0], tiles[wave][0][1], tile, lane);

  for (; tile < (long)numTiles; tile += stride) {
    const long nxt = tile + stride;
    if (nxt < (long)numTiles) {
      // prefetch next tile into the other buffer, then wait for current (30 newer
      // async ops may remain outstanding -> current tile's 30 are complete)
      issue_tile(gp, gl, lim, tiles[wave][buf ^ 1][0], tiles[wave][buf ^ 1][1], nxt, lane);
      WAIT_ASYNC(30);
    } else {
      WAIT_ASYNC(0);
    }
    asm volatile("" ::: "memory");

    const float* P = tiles[wave][buf][0];
    const float* L = tiles[wave][buf][1];
    const long c0 = tile * TILE_CELLS + lane;
    if (c0 < nCells)
      acc += cell_loss(P + lane * CELL_F, L + lane * CELL_F);
    if (c0 + 32 < nCells)
      acc += cell_loss(P + (lane + 32) * CELL_F, L + (lane + 32) * CELL_F);

    asm volatile("" ::: "memory");
    buf ^= 1;
  }

  // wave32 reduction
#pragma unroll
  for (int off = 16; off > 0; off >>= 1) acc += __shfl_down(acc, off, 32);
  if (lane == 0) wpart[wave] = acc;
  __syncthreads();
  if (threadIdx.x == 0) {
    float t = 0.0f;
#pragma unroll
    for (int w = 0; w < WAVES; ++w) t += wpart[w];
    blockPartials[blockIdx.x] = t;
  }
}

// ---------------- deterministic final reduction ----------------
__global__ void __launch_bounds__(256)
yolo_loss_final(const float* __restrict__ partials, int n, float* __restrict__ out,
                float scale) {
  __shared__ float red[8];
  float a = 0.0f;
  for (int i = threadIdx.x; i < n; i += blockDim.x) a += partials[i];
#pragma unroll
  for (int off = 16; off > 0; off >>= 1) a += __shfl_down(a, off, 32);
  const int wave = threadIdx.x >> 5, lane = threadIdx.x & 31;
  if (lane == 0) red[wave] = a;
  __syncthreads();
  if (threadIdx.x == 0) {
    float t = 0.0f;
#pragma unroll
    for (int w = 0; w < 8; ++w) t += red[w];
    *out = t * scale;
  }
}

// ---------------- host entry ----------------
extern "C" void kernel_launch(void* const* d_in, const int* in_sizes, int n_in,
                              void* d_out, int out_size, void* d_ws, size_t ws_size,
                              hipStream_t stream) {
  const float* preds  = (const float*)d_in[0];
  const float* labels = (const float*)d_in[1];
  float* out      = (float*)d_out;
  float* partials = (float*)d_ws;

  const long nCells   = (long)in_sizes[0] / CELL_F;        // N*S*S cells
  const int  numTiles = (int)((nCells + TILE_CELLS - 1) / TILE_CELLS);

  int blocks = 1024;                                       // 4096 wave32 streams
  const int maxBlocks = (numTiles + WAVES - 1) / WAVES;
  if (blocks > maxBlocks) blocks = (maxBlocks > 0) ? maxBlocks : 1;

  yolo_loss_main<<<blocks, 32 * WAVES, 0, stream>>>(preds, labels, partials,
                                                    nCells, numTiles);

  const float scale = (float)(49.0 / (double)nCells);      // == 1/N (batch)
  yolo_loss_final<<<1, 256, 0, stream>>>(partials, blocks, out, scale);
}